// GCN_2layer_48266842472557
// MI455X (gfx1250) — compile-verified
//
#include <hip/hip_runtime.h>
#include <hip/hip_bf16.h>

typedef float v2f __attribute__((ext_vector_type(2)));
typedef float v8f __attribute__((ext_vector_type(8)));

#define HCH 64   // hidden channels
#define CCH 16   // output classes
#define XPAD 68  // LDS row stride (floats): 272B = 16B-aligned, conflict-free banks

// ---------------------------------------------------------------------------
// 0) init: zero degree, pre-fold biases into accumulators
// ---------------------------------------------------------------------------
__global__ __launch_bounds__(256) void gcn_init_kernel(
    int* __restrict__ deg, float* __restrict__ out1, float* __restrict__ outF,
    const float* __restrict__ b1, const float* __restrict__ b2, int N) {
  long long gid = (long long)blockIdx.x * 256 + threadIdx.x;
  long long tot = (long long)N * HCH;
  if (gid < tot)                 out1[gid] = b1[gid & (HCH - 1)];
  if (gid < (long long)N * CCH)  outF[gid] = b2[gid & (CCH - 1)];
  if (gid < N)                   deg[gid] = 0;
}

// ---------------------------------------------------------------------------
// 1) degree over dst
// ---------------------------------------------------------------------------
__global__ __launch_bounds__(256) void gcn_degree_kernel(
    const int* __restrict__ dst, int* __restrict__ deg, int E) {
  int e = blockIdx.x * 256 + threadIdx.x;
  if (e < E) atomicAdd(&deg[dst[e]], 1);
}

// ---------------------------------------------------------------------------
// 2) dinv = rsqrt(deg + 1)   (self loop adds 1; deg >= 1 always)
// ---------------------------------------------------------------------------
__global__ __launch_bounds__(256) void gcn_dinv_kernel(
    const int* __restrict__ deg, float* __restrict__ dinv, int N) {
  int v = blockIdx.x * 256 + threadIdx.x;
  if (v < N) dinv[v] = rsqrtf((float)(deg[v] + 1));
}

// ---------------------------------------------------------------------------
// 3) GEMM1: h1 = x @ W1   [N,64] = [N,64]x[64,64], fp32 WMMA 16x16x4
//    block: 8 waves, 128 rows. Full tiles stage x via async global->LDS DMA
//    (GLOBAL_LOAD_ASYNC_TO_LDS_B128, ASYNCcnt); tail block uses guarded loads.
// ---------------------------------------------------------------------------
__global__ __launch_bounds__(256) void gcn_gemm1_kernel(
    const float* __restrict__ x, const float* __restrict__ W,
    float* __restrict__ h1, int N) {
  __shared__ float ldsX[128 * XPAD];
  __shared__ float ldsW[64 * 64];
  const int tid = threadIdx.x;
  const int rowBase = blockIdx.x * 128;

  { // W1: 4096 floats = 1024 float4
    const float4* Wv = (const float4*)W;
    float4* Lv = (float4*)ldsW;
    for (int i = tid; i < 1024; i += 256) Lv[i] = Wv[i];
  }

  if (rowBase + 128 <= N) {
    // Async DMA path: 2048 B128 transfers, 8 per thread, no VGPR round-trip.
    for (int i = tid; i < 128 * 16; i += 256) {
      int r = i >> 4, c4 = (i & 15) * 4;
      unsigned ldsAddr = (unsigned)(size_t)(&ldsX[r * XPAD + c4]); // 32-bit LDS byte offset
      unsigned gOff = (unsigned)(((long long)(rowBase + r)) * HCH + c4) * 4u;
      asm volatile("global_load_async_to_lds_b128 %0, %1, %2"
                   :: "v"(ldsAddr), "v"(gOff), "s"(x) : "memory");
    }
    asm volatile("s_wait_asynccnt 0" ::: "memory");
  } else {
    // Guarded synchronous path for the tail tile (zero-fill rows >= N).
    for (int i = tid; i < 128 * 16; i += 256) {
      int r = i >> 4, c4 = (i & 15) * 4;
      int gr = rowBase + r;
      float4 v = make_float4(0.f, 0.f, 0.f, 0.f);
      if (gr < N) v = *(const float4*)(x + (long long)gr * HCH + c4);
      float* p = &ldsX[r * XPAD + c4];
      p[0] = v.x; p[1] = v.y; p[2] = v.z; p[3] = v.w;
    }
  }
  __syncthreads();

  const int wave = tid >> 5, lane = tid & 31;
  const int m  = lane & 15;            // A row / B col / D col
  const int kh = (lane >> 4) * 2;      // K half-select
  const int wrow = wave * 16;
  const int rhalf = (lane >> 4) * 8;   // D row half-select

  for (int ct = 0; ct < 4; ++ct) {
    v8f acc = {};
    const int col0 = ct * 16;
    for (int kk = 0; kk < 16; ++kk) {
      const int k = kk * 4 + kh;
      v2f a, b;
      a.x = ldsX[(wrow + m) * XPAD + k];
      a.y = ldsX[(wrow + m) * XPAD + k + 1];
      b.x = ldsW[k * 64 + col0 + m];
      b.y = ldsW[(k + 1) * 64 + col0 + m];
      acc = __builtin_amdgcn_wmma_f32_16x16x4_f32(
          false, a, false, b, (short)0, acc, false, false);
    }
    for (int r = 0; r < 8; ++r) {
      int gr = rowBase + wrow + rhalf + r;
      if (gr < N) h1[(long long)gr * HCH + col0 + m] = acc[r];
    }
  }
}

// ---------------------------------------------------------------------------
// 4) scatter layer 1: wave per (edge or self-loop), lane = 2 channels
// ---------------------------------------------------------------------------
__global__ __launch_bounds__(256) void gcn_scatter1_kernel(
    const int* __restrict__ src, const int* __restrict__ dst,
    const float* __restrict__ dinv, const float* __restrict__ h,
    float* __restrict__ out1, int E, int total) {
  long long gid = (long long)blockIdx.x * 256 + threadIdx.x;
  int e = (int)(gid >> 5);
  if (e >= total) return;
  int lane = threadIdx.x & 31;
  int s, d;
  if (e < E) { s = src[e]; d = dst[e]; } else { s = e - E; d = s; }
  float nrm = dinv[s] * dinv[d];
  int c = lane * 2;
  float2 hv = *(const float2*)(h + (long long)s * HCH + c);
  atomicAdd(out1 + (long long)d * HCH + c,     hv.x * nrm);
  atomicAdd(out1 + (long long)d * HCH + c + 1, hv.y * nrm);
}

// ---------------------------------------------------------------------------
// 5) BN partial sums (deterministic two-stage reduction)
// ---------------------------------------------------------------------------
__global__ __launch_bounds__(256) void gcn_bn_partial_kernel(
    const float* __restrict__ out1, float* __restrict__ partial,
    int N, int rowsPerBlock) {
  const int c  = threadIdx.x & 63;
  const int rg = threadIdx.x >> 6;   // 0..3
  const int r0 = blockIdx.x * rowsPerBlock;
  const int r1 = min(N, r0 + rowsPerBlock);
  float s = 0.f, q = 0.f;
  for (int r = r0 + rg; r < r1; r += 4) {
    float v = out1[(long long)r * HCH + c];
    s += v; q += v * v;
  }
  __shared__ float ls[256], lq[256];
  ls[threadIdx.x] = s; lq[threadIdx.x] = q;
  __syncthreads();
  if (rg == 0) {
    s = ls[c] + ls[64 + c] + ls[128 + c] + ls[192 + c];
    q = lq[c] + lq[64 + c] + lq[128 + c] + lq[192 + c];
    partial[blockIdx.x * 128 + c]      = s;
    partial[blockIdx.x * 128 + 64 + c] = q;
  }
}

__global__ __launch_bounds__(64) void gcn_bn_finalize_kernel(
    const float* __restrict__ partial, const float* __restrict__ gamma,
    const float* __restrict__ beta, float* __restrict__ scaleshift,
    int nb, int N) {
  int c = threadIdx.x;            // 64 threads
  float s = 0.f, q = 0.f;
  for (int b = 0; b < nb; ++b) {  // fixed order -> deterministic
    s += partial[b * 128 + c];
    q += partial[b * 128 + 64 + c];
  }
  float invn = 1.0f / (float)N;
  float mean = s * invn;
  float var  = q * invn - mean * mean;           // biased, like reference
  float sc   = gamma[c] * rsqrtf(var + 1e-5f);
  scaleshift[c]      = sc;
  scaleshift[64 + c] = beta[c] - mean * sc;
}

// ---------------------------------------------------------------------------
// 6) GEMM2: h2 = relu(bn(out1)) @ W2   [N,16]; BN+ReLU fused into tile load
//    (transform in the staging path precludes DMA here)
// ---------------------------------------------------------------------------
__global__ __launch_bounds__(256) void gcn_gemm2_kernel(
    const float* __restrict__ out1, const float* __restrict__ W2,
    const float* __restrict__ scaleshift, float* __restrict__ h2, int N) {
  __shared__ float ldsX[128 * XPAD];
  __shared__ float ldsW[64 * 16];
  __shared__ float sscale[64], sshift[64];
  const int tid = threadIdx.x;
  const int rowBase = blockIdx.x * 128;

  if (tid < 64)  sscale[tid] = scaleshift[tid];
  else if (tid < 128) sshift[tid - 64] = scaleshift[tid];
  { // W2: 1024 floats = 256 float4
    const float4* Wv = (const float4*)W2;
    float4* Lv = (float4*)ldsW;
    if (tid < 256) Lv[tid] = Wv[tid];
  }
  __syncthreads();   // sscale/sshift ready before fused tile load

  for (int i = tid; i < 128 * 16; i += 256) {
    int r = i >> 4, c4 = (i & 15) * 4;
    int gr = rowBase + r;
    float4 v = make_float4(0.f, 0.f, 0.f, 0.f);
    if (gr < N) v = *(const float4*)(out1 + (long long)gr * HCH + c4);
    float* p = &ldsX[r * XPAD + c4];
    p[0] = fmaxf(0.f, v.x * sscale[c4]     + sshift[c4]);
    p[1] = fmaxf(0.f, v.y * sscale[c4 + 1] + sshift[c4 + 1]);
    p[2] = fmaxf(0.f, v.z * sscale[c4 + 2] + sshift[c4 + 2]);
    p[3] = fmaxf(0.f, v.w * sscale[c4 + 3] + sshift[c4 + 3]);
  }
  __syncthreads();

  const int wave = tid >> 5, lane = tid & 31;
  const int m  = lane & 15;
  const int kh = (lane >> 4) * 2;
  const int wrow = wave * 16;
  const int rhalf = (lane >> 4) * 8;

  v8f acc = {};
  for (int kk = 0; kk < 16; ++kk) {
    const int k = kk * 4 + kh;
    v2f a, b;
    a.x = ldsX[(wrow + m) * XPAD + k];
    a.y = ldsX[(wrow + m) * XPAD + k + 1];
    b.x = ldsW[k * CCH + m];
    b.y = ldsW[(k + 1) * CCH + m];
    acc = __builtin_amdgcn_wmma_f32_16x16x4_f32(
        false, a, false, b, (short)0, acc, false, false);
  }
  for (int r = 0; r < 8; ++r) {
    int gr = rowBase + wrow + rhalf + r;
    if (gr < N) h2[(long long)gr * CCH + m] = acc[r];
  }
}

// ---------------------------------------------------------------------------
// 7) scatter layer 2: thread per (edge, channel), 16 channels
// ---------------------------------------------------------------------------
__global__ __launch_bounds__(256) void gcn_scatter2_kernel(
    const int* __restrict__ src, const int* __restrict__ dst,
    const float* __restrict__ dinv, const float* __restrict__ h2,
    float* __restrict__ outF, int E, int total) {
  long long gid = (long long)blockIdx.x * 256 + threadIdx.x;
  int e = (int)(gid >> 4);
  if (e >= total) return;
  int c = (int)(gid & 15);
  int s, d;
  if (e < E) { s = src[e]; d = dst[e]; } else { s = e - E; d = s; }
  float nrm = dinv[s] * dinv[d];
  atomicAdd(outF + (long long)d * CCH + c, h2[(long long)s * CCH + c] * nrm);
}

// ---------------------------------------------------------------------------
extern "C" void kernel_launch(void* const* d_in, const int* in_sizes, int n_in,
                              void* d_out, int out_size, void* d_ws, size_t ws_size,
                              hipStream_t stream) {
  const float* x     = (const float*)d_in[0];
  const int*   eidx  = (const int*)d_in[1];
  const float* W1    = (const float*)d_in[2];
  const float* b1    = (const float*)d_in[3];
  const float* gamma = (const float*)d_in[4];
  const float* beta  = (const float*)d_in[5];
  const float* W2    = (const float*)d_in[6];
  const float* b2    = (const float*)d_in[7];
  float* outF = (float*)d_out;

  const int N = in_sizes[0] / HCH;
  const int E = in_sizes[1] / 2;
  const int total = E + N;
  const int* srcp = eidx;
  const int* dstp = eidx + E;

  // workspace carve-up (256B aligned)
  char* base = (char*)d_ws;
  size_t off = 0;
  auto carve = [&](size_t bytes) { char* p = base + off; off = (off + bytes + 255) & ~(size_t)255; return p; };
  int*   deg        = (int*)  carve((size_t)N * 4);
  float* dinv       = (float*)carve((size_t)N * 4);
  float* h1         = (float*)carve((size_t)N * HCH * 4); // reused as h2 [N,16]
  float* out1       = (float*)carve((size_t)N * HCH * 4);
  const int NB = 200;
  float* partial    = (float*)carve((size_t)NB * 128 * 4);
  float* scaleshift = (float*)carve(128 * 4);
  float* h2 = h1;   // h1 dead after scatter1

  const long long initN = (long long)N * HCH;
  gcn_init_kernel<<<(int)((initN + 255) / 256), 256, 0, stream>>>(deg, out1, outF, b1, b2, N);
  gcn_degree_kernel<<<(E + 255) / 256, 256, 0, stream>>>(dstp, deg, E);
  gcn_dinv_kernel<<<(N + 255) / 256, 256, 0, stream>>>(deg, dinv, N);
  gcn_gemm1_kernel<<<(N + 127) / 128, 256, 0, stream>>>(x, W1, h1, N);
  {
    long long work = (long long)total * 32;
    gcn_scatter1_kernel<<<(int)((work + 255) / 256), 256, 0, stream>>>(
        srcp, dstp, dinv, h1, out1, E, total);
  }
  const int rowsPerBlock = (N + NB - 1) / NB;
  gcn_bn_partial_kernel<<<NB, 256, 0, stream>>>(out1, partial, N, rowsPerBlock);
  gcn_bn_finalize_kernel<<<1, 64, 0, stream>>>(partial, gamma, beta, scaleshift, NB, N);
  gcn_gemm2_kernel<<<(N + 127) / 128, 256, 0, stream>>>(out1, W2, scaleshift, h2, N);
  {
    long long work = (long long)total * 16;
    gcn_scatter2_kernel<<<(int)((work + 255) / 256), 256, 0, stream>>>(
        srcp, dstp, dinv, h2, outF, E, total);
  }
}